// GraphAttentionLayerEdge_13005160972434
// MI455X (gfx1250) — compile-verified
//
#include <hip/hip_runtime.h>
#include <hip/hip_bf16.h>

#define N_NODES 50000
#define N_EDGES 800000
#define D_NODE 128
#define D_EDGE 64
#define D_HID 64
#define SLOPE 0.2f

typedef float v2f __attribute__((ext_vector_type(2)));
typedef float v8f __attribute__((ext_vector_type(8)));

// ---------------------------------------------------------------------------
// Wave-level helpers (wave32 on gfx1250)
// ---------------------------------------------------------------------------
__device__ inline float wave_reduce_sum(float v) {
#pragma unroll
    for (int off = 16; off > 0; off >>= 1)
        v += __shfl_down(v, off, 32);
    return v;
}

// float atomic max via signed/unsigned int trick (works for mixed signs,
// init value must be -inf = 0xFF800000)
__device__ inline void atomicMaxFloat(float* addr, float val) {
    if (val >= 0.0f)
        atomicMax((int*)addr, __float_as_int(val));
    else
        atomicMin((unsigned int*)addr, __float_as_uint(val));
}

// ---------------------------------------------------------------------------
// WMMA GEMM:  C[M x 64] = A[M x K] @ W[64 x K]^T + bias[64]
// One wave computes a 16x64 tile: 4 accumulators, A-fragment reused.
// Uses V_WMMA_F32_16X16X4_F32 (full fp32, matches reference precision).
//
// A frag (16x4, MxK): lanes 0-15 -> M=lane, K={k0,k0+1}; lanes 16-31 -> K={k0+2,k0+3}
// B frag (4x16, KxN): lanes 0-15 -> N=lane, K={k0,k0+1}; lanes 16-31 -> K={k0+2,k0+3}
// C tile  (16x16):    VGPR v, lanes 0-15 -> M=v, N=lane; lanes 16-31 -> M=v+8
// ---------------------------------------------------------------------------
template <int K>
__global__ __launch_bounds__(256) void wmma_gemm_bias(
        const float* __restrict__ A, const float* __restrict__ W,
        const float* __restrict__ bias, float* __restrict__ C, int M) {
    const int lane  = threadIdx.x & 31;
    const int wave  = threadIdx.x >> 5;
    const int mtile = blockIdx.x * 8 + wave;
    if (mtile * 16 >= M) return;                 // wave-uniform: EXEC stays full

    const int mrow  = (lane & 15);               // M within tile for A frag
    const int ncol  = (lane & 15);               // N within tile for B/C frags
    const int khalf = (lane >> 4) << 1;          // 0 or 2

    const float* arow = A + (size_t)(mtile * 16 + mrow) * K;

    v8f c0 = {}, c1 = {}, c2 = {}, c3 = {};

#pragma unroll 4
    for (int k0 = 0; k0 < K; k0 += 4) {
        v2f a = *(const v2f*)(arow + k0 + khalf);
        // B[k][n] = W[n][k]; W row-major (64 x K), consecutive k -> v2f load
        v2f b0 = *(const v2f*)(W + (size_t)(ncol +  0) * K + k0 + khalf);
        v2f b1 = *(const v2f*)(W + (size_t)(ncol + 16) * K + k0 + khalf);
        v2f b2 = *(const v2f*)(W + (size_t)(ncol + 32) * K + k0 + khalf);
        v2f b3 = *(const v2f*)(W + (size_t)(ncol + 48) * K + k0 + khalf);
        c0 = __builtin_amdgcn_wmma_f32_16x16x4_f32(false, a, false, b0, (short)0, c0, false, false);
        c1 = __builtin_amdgcn_wmma_f32_16x16x4_f32(false, a, false, b1, (short)0, c1, false, false);
        c2 = __builtin_amdgcn_wmma_f32_16x16x4_f32(false, a, false, b2, (short)0, c2, false, false);
        c3 = __builtin_amdgcn_wmma_f32_16x16x4_f32(false, a, false, b3, (short)0, c3, false, false);
    }

    const int rbase = mtile * 16 + ((lane >> 4) << 3);
#pragma unroll
    for (int v = 0; v < 8; ++v) {
        float* crow = C + (size_t)(rbase + v) * D_HID;
        crow[ 0 + ncol] = c0[v] + bias[ 0 + ncol];
        crow[16 + ncol] = c1[v] + bias[16 + ncol];
        crow[32 + ncol] = c2[v] + bias[32 + ncol];
        crow[48 + ncol] = c3[v] + bias[48 + ncol];
    }
}

// ---------------------------------------------------------------------------
// init m = -inf, z = 0
// ---------------------------------------------------------------------------
__global__ void init_mz_kernel(float* __restrict__ m, float* __restrict__ z) {
    int n = blockIdx.x * blockDim.x + threadIdx.x;
    if (n < N_NODES) {
        m[n] = __int_as_float(0xFF800000);  // -inf
        z[n] = 0.0f;
    }
}

// ---------------------------------------------------------------------------
// s_i[n] = h[n] . a[0:64],  s_j[n] = h[n] . a[64:128]   (wave per node)
// ---------------------------------------------------------------------------
__global__ __launch_bounds__(256) void node_score_kernel(
        const float* __restrict__ h, const float* __restrict__ a,
        float* __restrict__ s_i, float* __restrict__ s_j) {
    const int lane = threadIdx.x & 31;
    const int n    = blockIdx.x * 8 + (threadIdx.x >> 5);
    if (n >= N_NODES) return;
    const float* hr = h + (size_t)n * D_HID;
    float h0 = hr[lane], h1 = hr[lane + 32];
    float pi = h0 * a[lane]         + h1 * a[lane + 32];
    float pj = h0 * a[D_HID + lane] + h1 * a[D_HID + lane + 32];
    pi = wave_reduce_sum(pi);
    pj = wave_reduce_sum(pj);
    if (lane == 0) { s_i[n] = pi; s_j[n] = pj; }
}

// ---------------------------------------------------------------------------
// eij = leaky_relu(s_i[tgt] + s_j[nbr] + edge_h[e] . a_e); segment max into m
// (wave per edge)
// ---------------------------------------------------------------------------
__global__ __launch_bounds__(256) void edge_score_kernel(
        const float* __restrict__ edge_h, const float* __restrict__ a,
        const int* __restrict__ tgt, const int* __restrict__ nbr,
        const float* __restrict__ s_i, const float* __restrict__ s_j,
        float* __restrict__ eij, float* __restrict__ m) {
    const int lane = threadIdx.x & 31;
    const int e    = blockIdx.x * 8 + (threadIdx.x >> 5);
    if (e >= N_EDGES) return;
    const float* eh = edge_h + (size_t)e * D_HID;
    const float* ae = a + 2 * D_HID;
    float p = eh[lane] * ae[lane] + eh[lane + 32] * ae[lane + 32];
    p = wave_reduce_sum(p);
    if (lane == 0) {
        const int t = tgt[e];
        float x = s_i[t] + s_j[nbr[e]] + p;
        x = (x > 0.0f) ? x : SLOPE * x;
        eij[e] = x;
        atomicMaxFloat(m + t, x);
    }
}

// ---------------------------------------------------------------------------
// ex = exp(eij - m[tgt]); z[tgt] += ex   (thread per edge; in-place on eij)
// ---------------------------------------------------------------------------
__global__ void exp_sum_kernel(const int* __restrict__ tgt,
                               const float* __restrict__ m,
                               float* __restrict__ ev,
                               float* __restrict__ z) {
    int e = blockIdx.x * blockDim.x + threadIdx.x;
    if (e < N_EDGES) {
        const int t = tgt[e];
        float ex = expf(ev[e] - m[t]);
        ev[e] = ex;
        atomicAdd(z + t, ex);
    }
}

// ---------------------------------------------------------------------------
// out = h   (residual term, also the accumulation base)
// ---------------------------------------------------------------------------
__global__ void copy_kernel(const float* __restrict__ src, float* __restrict__ dst, int n) {
    int i = blockIdx.x * blockDim.x + threadIdx.x;
    if (i < n) dst[i] = src[i];
}

// ---------------------------------------------------------------------------
// out[tgt] += (ex/z[tgt]) * h[nbr]   (wave per edge, 2 channels per lane)
// ---------------------------------------------------------------------------
__global__ __launch_bounds__(256) void aggregate_kernel(
        const int* __restrict__ tgt, const int* __restrict__ nbr,
        const float* __restrict__ ev, const float* __restrict__ z,
        const float* __restrict__ h, float* __restrict__ out) {
    const int lane = threadIdx.x & 31;
    const int e    = blockIdx.x * 8 + (threadIdx.x >> 5);
    if (e >= N_EDGES) return;
    const int t  = tgt[e];
    const int nb = nbr[e];
    const float alpha = ev[e] / z[t];
    const float* hn = h + (size_t)nb * D_HID;
    float* orow = out + (size_t)t * D_HID;
    atomicAdd(orow + lane,      alpha * hn[lane]);
    atomicAdd(orow + lane + 32, alpha * hn[lane + 32]);
}

// ---------------------------------------------------------------------------
extern "C" void kernel_launch(void* const* d_in, const int* in_sizes, int n_in,
                              void* d_out, int out_size, void* d_ws, size_t ws_size,
                              hipStream_t stream) {
    const float* node_features = (const float*)d_in[0];
    const float* edge_features = (const float*)d_in[1];
    const int*   edge_index    = (const int*)d_in[2];
    const float* Ww            = (const float*)d_in[3];
    const float* bw            = (const float*)d_in[4];
    const float* We            = (const float*)d_in[5];
    const float* be            = (const float*)d_in[6];
    const float* a             = (const float*)d_in[7];

    const int* tgt = edge_index;             // edge_index[0]
    const int* nbr = edge_index + N_EDGES;   // edge_index[1]

    float* out0   = (float*)d_out;                          // (N_NODES, 64)
    float* edge_h = out0 + (size_t)N_NODES * D_HID;         // (N_EDGES, 64)

    float* ws_f = (float*)d_ws;
    float* h    = ws_f;                                     // N_NODES*64
    float* s_i  = h + (size_t)N_NODES * D_HID;              // N_NODES
    float* s_j  = s_i + N_NODES;                            // N_NODES
    float* m    = s_j + N_NODES;                            // N_NODES
    float* z    = m + N_NODES;                              // N_NODES
    float* ev   = z + N_NODES;                              // N_EDGES

    // 1) h = node_features @ Ww^T + bw   (3125 M-tiles -> 391 blocks of 8 waves)
    wmma_gemm_bias<D_NODE><<<(N_NODES / 16 + 7) / 8, 256, 0, stream>>>(
        node_features, Ww, bw, h, N_NODES);

    // 2) edge_h = edge_features @ We^T + be   (50000 M-tiles -> 6250 blocks)
    wmma_gemm_bias<D_EDGE><<<(N_EDGES / 16 + 7) / 8, 256, 0, stream>>>(
        edge_features, We, be, edge_h, N_EDGES);

    // 3) m = -inf, z = 0
    init_mz_kernel<<<(N_NODES + 255) / 256, 256, 0, stream>>>(m, z);

    // 4) s_i, s_j
    node_score_kernel<<<(N_NODES + 7) / 8, 256, 0, stream>>>(h, a, s_i, s_j);

    // 5) eij + segment max
    edge_score_kernel<<<(N_EDGES + 7) / 8, 256, 0, stream>>>(
        edge_h, a, tgt, nbr, s_i, s_j, ev, m);

    // 6) ex + segment sum
    exp_sum_kernel<<<(N_EDGES + 255) / 256, 256, 0, stream>>>(tgt, m, ev, z);

    // 7) out = h (residual base)
    copy_kernel<<<((N_NODES * D_HID) + 255) / 256, 256, 0, stream>>>(
        h, out0, N_NODES * D_HID);

    // 8) out[tgt] += alpha * h[nbr]
    aggregate_kernel<<<(N_EDGES + 7) / 8, 256, 0, stream>>>(
        tgt, nbr, ev, z, h, out0);
}